// GNN_13529146982743
// MI455X (gfx1250) — compile-verified
//
#include <hip/hip_runtime.h>
#include <hip/hip_bf16.h>

typedef __attribute__((ext_vector_type(2))) float v2f;
typedef __attribute__((ext_vector_type(8))) float v8f;

#define D 64

// ---------------------------------------------------------------- zero fill
__global__ __launch_bounds__(256)
void gnn_zero_f32(float* __restrict__ p, int n) {
    int i = blockIdx.x * blockDim.x + threadIdx.x;
    float4 z = make_float4(0.f, 0.f, 0.f, 0.f);
    if (i * 4 < n) ((float4*)p)[i] = z;
}

// ------------------------------------------------- edge scatter (segment sum)
// 16 lanes per edge, float4 gather + 4 native f32 atomics into agg[dst].
__global__ __launch_bounds__(256)
void gnn_scatter_add(const float* __restrict__ x,
                     const long long* __restrict__ ei,   // [2, n_edges]
                     float* __restrict__ agg, int n_edges) {
    long long t = (long long)blockIdx.x * blockDim.x + threadIdx.x;
    long long e = t >> 4;
    int c = (int)(t & 15) * 4;
    if (e >= n_edges) return;
    long long src = ei[e];
    long long dst = ei[(long long)n_edges + e];
    const float4 v = *(const float4*)(x + src * D + c);
    float* p = agg + dst * D + c;
    unsafeAtomicAdd(p + 0, v.x);
    unsafeAtomicAdd(p + 1, v.y);
    unsafeAtomicAdd(p + 2, v.z);
    unsafeAtomicAdd(p + 3, v.w);
}

// --------------------------------------------- fused graph-conv GEMM (WMMA)
// out[16 rows x 64 cols] = X@Wr + AGG@Wn + b  (optional ReLU)
// One wave32 per 16-node tile; V_WMMA_F32_16X16X4_F32, K-loop of 16 steps.
template <bool RELU>
__global__ __launch_bounds__(128)
void gnn_gconv_gemm(const float* __restrict__ X, const float* __restrict__ AGG,
                    const float* __restrict__ Wr, const float* __restrict__ Wn,
                    const float* __restrict__ bias, float* __restrict__ out,
                    int n_nodes) {
    const int lane = threadIdx.x & 31;
    const int wave = threadIdx.x >> 5;
    const int row0 = (blockIdx.x * 4 + wave) * 16;
    if (row0 >= n_nodes) return;

    const int m  = lane & 15;          // row within tile (A), col within tile (B/C)
    const int kh = (lane >> 4) * 2;    // lane-half K offset: 0 or 2

    // clamp A-row reads for (non-existent here) partial tiles
    int arow_i = row0 + m;
    if (arow_i >= n_nodes) arow_i = n_nodes - 1;
    const float* xrow = X   + (long long)arow_i * D;
    const float* grow = AGG + (long long)arow_i * D;

    // C/D layout: bias depends only on N = t*16 + (lane&15), same for all 8 VGPRs
    v8f acc[4];
#pragma unroll
    for (int t = 0; t < 4; ++t) {
        float bv = bias[t * 16 + m];
#pragma unroll
        for (int i = 0; i < 8; ++i) acc[t][i] = bv;
    }

#pragma unroll
    for (int k = 0; k < 16; ++k) {
        const int kk = k * 4 + kh;                       // this lane's K pair base
        v2f ax = *(const v2f*)(xrow + kk);               // A frag (x)
        v2f ag = *(const v2f*)(grow + kk);               // A frag (agg)
#pragma unroll
        for (int t = 0; t < 4; ++t) {
            const int n = t * 16 + m;
            v2f br = { Wr[kk * D + n], Wr[(kk + 1) * D + n] };   // B frag (W_root)
            v2f bn = { Wn[kk * D + n], Wn[(kk + 1) * D + n] };   // B frag (W_nbr)
            acc[t] = __builtin_amdgcn_wmma_f32_16x16x4_f32(
                false, ax, false, br, (short)0, acc[t], false, false);
            acc[t] = __builtin_amdgcn_wmma_f32_16x16x4_f32(
                false, ag, false, bn, (short)0, acc[t], false, false);
        }
    }

    // store: VGPR r -> M = r (lanes 0-15) or r+8 (lanes 16-31), N = t*16 + m
    const int mh = (lane >> 4) * 8;
#pragma unroll
    for (int t = 0; t < 4; ++t) {
        const int n = t * 16 + m;
#pragma unroll
        for (int r = 0; r < 8; ++r) {
            int mrow = row0 + mh + r;
            if (mrow < n_nodes) {
                float v = acc[t][r];
                if (RELU) v = v > 0.f ? v : 0.f;
                out[(long long)mrow * D + n] = v;
            }
        }
    }
}

// ----------------------------------------------------------------- launcher
extern "C" void kernel_launch(void* const* d_in, const int* in_sizes, int n_in,
                              void* d_out, int out_size, void* d_ws, size_t ws_size,
                              hipStream_t stream) {
    const float*     x   = (const float*)d_in[0];
    const long long* ei  = (const long long*)d_in[1];
    const float*     W1r = (const float*)d_in[2];
    const float*     W1n = (const float*)d_in[3];
    const float*     b1  = (const float*)d_in[4];
    const float*     W2r = (const float*)d_in[5];
    const float*     W2n = (const float*)d_in[6];
    const float*     b2  = (const float*)d_in[7];
    float*           out = (float*)d_out;

    const int n_nodes = in_sizes[0] / D;
    const int n_edges = in_sizes[1] / 2;

    float* agg = (float*)d_ws;                       // [n_nodes, 64]
    float* h   = agg + (size_t)n_nodes * D;          // [n_nodes, 64]

    const int feat = n_nodes * D;
    dim3 zblk(256), zgrd((feat / 4 + 255) / 256);
    dim3 sblk(256), sgrd((unsigned)(((long long)n_edges * 16 + 255) / 256));
    const int tiles = (n_nodes + 15) / 16;
    dim3 gblk(128), ggrd((tiles + 3) / 4);

    // layer 1
    gnn_zero_f32<<<zgrd, zblk, 0, stream>>>(agg, feat);
    gnn_scatter_add<<<sgrd, sblk, 0, stream>>>(x, ei, agg, n_edges);
    gnn_gconv_gemm<true><<<ggrd, gblk, 0, stream>>>(x, agg, W1r, W1n, b1, h, n_nodes);

    // layer 2
    gnn_zero_f32<<<zgrd, zblk, 0, stream>>>(agg, feat);
    gnn_scatter_add<<<sgrd, sblk, 0, stream>>>(h, ei, agg, n_edges);
    gnn_gconv_gemm<false><<<ggrd, gblk, 0, stream>>>(h, agg, W2r, W2n, b2, out, n_nodes);
}